// GCNLayer_22892175687923
// MI455X (gfx1250) — compile-verified
//
#include <hip/hip_runtime.h>

typedef __attribute__((ext_vector_type(2))) float v2f;
typedef __attribute__((ext_vector_type(8))) float v8f;

#define DIM 64
#define LEAKY 0.01f
// Packed-W LDS row stride in float2 units: 80*8B = 640B = 160 DWORDs == 32 mod 64
// -> consecutive kp rows land on opposite bank halves => conflict-free b64 loads.
#define WSTRIDE 80

// ---------------- Phase 0: zero the Gamma accumulator (workspace) -----------
__global__ void gcn_zero_kernel(float4* __restrict__ g, int n4) {
    int i = blockIdx.x * blockDim.x + threadIdx.x;
    int stride = gridDim.x * blockDim.x;
    float4 z; z.x = 0.f; z.y = 0.f; z.z = 0.f; z.w = 0.f;
    for (; i < n4; i += stride) g[i] = z;
}

// ---------------- Phase 1: scatter-add H_u rows into Gamma[V[e]] ------------
// One thread per (edge, 4-float group): float4 load + 4 f32 atomics.
// Gamma is 12.8 MB -> atomics resolve in L2 (192 MB); HBM sees only the
// 204.8 MB H_u stream.
__global__ void gcn_scatter_kernel(const float* __restrict__ Hu,
                                   const int* __restrict__ V,
                                   float* __restrict__ Gamma,
                                   int E) {
    int t = blockIdx.x * blockDim.x + threadIdx.x;
    int total = E * (DIM / 4);
    int stride = gridDim.x * blockDim.x;
    for (; t < total; t += stride) {
        int e = t >> 4;          // edge index (DIM/4 == 16 groups per edge)
        int q = t & 15;          // which float4 within the row
        const float4 h = *reinterpret_cast<const float4*>(Hu + (size_t)e * DIM + q * 4);
        int v = V[e];
        float* dst = Gamma + (size_t)v * DIM + q * 4;
        atomicAdd(dst + 0, h.x);
        atomicAdd(dst + 1, h.y);
        atomicAdd(dst + 2, h.z);
        atomicAdd(dst + 3, h.w);
    }
}

// ---------------- Phase 2: out = LeakyReLU(Gamma @ W + b) via f32 WMMA ------
// Each wave computes a 16x64 output tile: 4 accumulators (16x16 each),
// K-loop over 64 in steps of 4 using V_WMMA_F32_16X16X4_F32.
// A-fragment (16x4 f32, 2 VGPR): lane l -> row l&15, K = {2*(l>>4), 2*(l>>4)+1}
// B-fragment (4x16 f32, 2 VGPR): lane l -> col l&15, same K pair
// C/D (16x16 f32, 8 VGPR): vgpr g -> row g + 8*(l>>4), col l&15
//
// W is pre-packed into LDS as float2 {W[2kp][col], W[2kp+1][col]} so each
// B fragment is ONE contiguous ds_load_b64 (no register shuffling).
__global__ void __launch_bounds__(256)
gcn_gemm_kernel(const float* __restrict__ Gamma,
                const float* __restrict__ W,
                const float* __restrict__ bias,
                float* __restrict__ out,
                int nRowTiles, int nRows) {
    __shared__ v2f Wlds[(DIM / 2) * WSTRIDE];   // 20 KB of the 320 KB WGP LDS

    // Cooperative pack: Wlds[kp*WSTRIDE + col] = {W[2kp][col], W[2kp+1][col]}
    for (int i = threadIdx.x; i < (DIM / 2) * DIM; i += blockDim.x) {
        int kp  = i >> 6;       // K-pair index 0..31
        int col = i & 63;
        v2f w;
        w.x = W[(2 * kp) * DIM + col];
        w.y = W[(2 * kp + 1) * DIM + col];
        Wlds[kp * WSTRIDE + col] = w;
    }
    __syncthreads();

    const int lane = threadIdx.x & 31;
    const int wave = threadIdx.x >> 5;
    const int tile = blockIdx.x * 8 + wave;    // 8 wave32 per 256-thread block
    if (tile >= nRowTiles) return;             // wave-uniform: EXEC stays all-1s

    const int row     = lane & 15;
    const int khalf   = lane >> 4;             // 0 or 1
    const int rowbase = tile * 16;
    // clamp load row for a possible ragged last tile (N=50000 is exact anyway)
    const int arowIdx = (rowbase + row < nRows) ? (rowbase + row) : (nRows - 1);
    const float* __restrict__ arow = Gamma + (size_t)arowIdx * DIM;

    v8f acc0 = {}, acc1 = {}, acc2 = {}, acc3 = {};

    #pragma unroll
    for (int k = 0; k < DIM; k += 4) {
        const int kp = (k >> 1) + khalf;       // this half-wave's K-pair row
        // A fragment: contiguous f32 pair -> one global_load_b64
        v2f a = *reinterpret_cast<const v2f*>(arow + 2 * kp);

        // B fragments: one conflict-free ds_load_b64 each
        const v2f* __restrict__ brow = &Wlds[kp * WSTRIDE + row];
        v2f b0 = brow[0];
        v2f b1 = brow[16];
        v2f b2 = brow[32];
        v2f b3 = brow[48];

        acc0 = __builtin_amdgcn_wmma_f32_16x16x4_f32(false, a, false, b0, (short)0, acc0, false, false);
        acc1 = __builtin_amdgcn_wmma_f32_16x16x4_f32(false, a, false, b1, (short)0, acc1, false, false);
        acc2 = __builtin_amdgcn_wmma_f32_16x16x4_f32(false, a, false, b2, (short)0, acc2, false, false);
        acc3 = __builtin_amdgcn_wmma_f32_16x16x4_f32(false, a, false, b3, (short)0, acc3, false, false);
    }

    // Epilogue: bias + LeakyReLU, write D tiles (coalesced b32 stores)
    const int mhigh = khalf * 8;
    #pragma unroll
    for (int ct = 0; ct < 4; ++ct) {
        const v8f a = (ct == 0) ? acc0 : (ct == 1) ? acc1 : (ct == 2) ? acc2 : acc3;
        const int col = ct * 16 + row;
        const float bb = bias[col];
        #pragma unroll
        for (int g = 0; g < 8; ++g) {
            const int r = rowbase + g + mhigh;
            if (r < nRows) {
                float v = a[g] + bb;
                v = (v >= 0.f) ? v : LEAKY * v;
                out[(size_t)r * DIM + col] = v;
            }
        }
    }
}

// ---------------------------------------------------------------------------
extern "C" void kernel_launch(void* const* d_in, const int* in_sizes, int n_in,
                              void* d_out, int out_size, void* d_ws, size_t ws_size,
                              hipStream_t stream) {
    // setup_inputs order:
    // 0:H_v(1) 1:H_u(E*64) 2:X_e(1) 3:batch_indices(N) 4:V(E) 5:U(E)
    // 6:X_v(1) 7:W(64*64) 8:b(64)
    const float* Hu    = (const float*)d_in[1];
    const int*   V     = (const int*)d_in[4];
    const float* W     = (const float*)d_in[7];
    const float* bias  = (const float*)d_in[8];
    float*       out   = (float*)d_out;
    float*       Gamma = (float*)d_ws;   // N*64 f32 = 12.8 MB scratch

    const int E = in_sizes[1] / DIM;
    const int N = in_sizes[3];

    // Phase 0: zero Gamma
    const int n4 = N * (DIM / 4);
    gcn_zero_kernel<<<(n4 + 255) / 256, 256, 0, stream>>>(
        reinterpret_cast<float4*>(Gamma), n4);

    // Phase 1: scatter-add (bandwidth-bound: 204.8 MB stream of H_u)
    const int scatterWork = E * (DIM / 4);
    int scatterBlocks = (scatterWork + 255) / 256;
    if (scatterBlocks > 65535) scatterBlocks = 65535;  // grid-stride covers rest
    gcn_scatter_kernel<<<scatterBlocks, 256, 0, stream>>>(Hu, V, Gamma, E);

    // Phase 2: WMMA GEMM + bias + LeakyReLU
    const int nRowTiles = (N + 15) / 16;
    gcn_gemm_kernel<<<(nRowTiles + 7) / 8, 256, 0, stream>>>(
        Gamma, W, bias, out, nRowTiles, N);
}